// GPUAugmentationPipeline_19636590477513
// MI455X (gfx1250) — compile-verified
//
#include <hip/hip_runtime.h>
#include <hip/hip_bf16.h>
#include <stdint.h>

// ---------- problem constants ----------
#define BB 16
#define CC 3
#define TT 16
#define HH 224
#define WW 224
#define COLOR_STRENGTH 0.3f
#define NOISE_STRENGTH 0.02f
#define DROPOUT_PROB 0.05f

constexpr uint32_t W4      = WW / 4;              // 56 float4 per row
constexpr uint32_t SLICE4  = HH * W4;             // 12544 float4 per (b,c,t) slice
constexpr uint32_t SLICES  = BB * CC * TT;        // 768 slices
constexpr uint32_t BLK_X   = SLICE4 / 256;        // 49 blocks of 256 threads per slice

typedef float v4f __attribute__((ext_vector_type(4)));

struct AugParams {
    float bf[BB];          // brightness factor
    float cf[BB];          // contrast factor (clamped)
    int   jt[BB];          // jitter shift j in [-3,3]
    int   flip[BB];        // 1 = horizontal flip
    float keep[BB * TT];   // frame-dropout mask (with >=1-kept fixup)
};

// ---------- prolog: fold all per-sample randomness into a small param block ----------
__global__ void aug_params_kernel(const int*   __restrict__ jitter,
                                  const float* __restrict__ b_rand,
                                  const float* __restrict__ c_rand,
                                  const float* __restrict__ flip_rand,
                                  const float* __restrict__ drop_rand,
                                  AugParams*   __restrict__ p) {
    int b = threadIdx.x;
    if (b < BB) {
        p->bf[b] = 1.0f + (b_rand[b] - 0.5f) * 2.0f * COLOR_STRENGTH;
        float cf = 1.0f + (c_rand[b] - 0.5f) * 2.0f * COLOR_STRENGTH;
        p->cf[b] = fminf(fmaxf(cf, 0.1f), 3.0f);
        p->jt[b] = jitter[b] - 3;                // MAX_JITTER = 3
        p->flip[b] = (flip_rand[b] < 0.5f) ? 1 : 0;

        bool k[TT];
        bool any = false;
        #pragma unroll
        for (int t = 0; t < TT; ++t) {
            k[t] = drop_rand[b * TT + t] > DROPOUT_PROB;
            any |= k[t];
        }
        if (!any) k[0] = true;                   // guarantee >=1 kept frame
        #pragma unroll
        for (int t = 0; t < TT; ++t)
            p->keep[b * TT + t] = k[t] ? 1.0f : 0.0f;
    }
}

__device__ __forceinline__ float clamp01(float x) {
    return fminf(fmaxf(x, 0.0f), 1.0f);
}

// ---------- main streaming kernel ----------
// grid = (49, 768): blockIdx.y enumerates (b,c,t) slices -> all per-sample
// parameters are block-uniform and compile to SGPR loads / SALU; each thread
// streams exactly one float4 with NT b128 traffic.
__global__ __launch_bounds__(256) void aug_main_kernel(
        const float* __restrict__ video,
        const float* __restrict__ noise,
        const AugParams* __restrict__ p,
        float* __restrict__ out) {
    // --- uniform (scalar) part ---
    uint32_t s  = blockIdx.y;          // slice = (b*CC + c)*TT + t
    uint32_t t  = s % TT;
    uint32_t bc = s / TT;              // b*CC + c
    uint32_t b  = bc / CC;

    const int   j    = p->jt[b];
    const int   flip = p->flip[b];
    const float bf   = p->bf[b];
    const float cf   = p->cf[b];
    const float kp   = p->keep[b * TT + t];     // dropout uses OUTPUT t

    // temporal roll: src frame st = (t - j) mod 16
    uint32_t st      = (uint32_t)(((int)t - j + TT) & (TT - 1));
    uint32_t vslice  = bc * TT + st;            // video source slice

    // --- per-thread part ---
    uint32_t pos = blockIdx.x * 256u + threadIdx.x;   // [0, 12544)
    uint32_t w4  = pos % W4;
    uint32_t h   = pos / W4;

    // flip happens AFTER noise-add in the reference, so both video and noise
    // sample the mirrored column: out[w] <- src[W-1-w]
    uint32_t src_pos = flip ? (h * W4 + (W4 - 1u - w4)) : pos;

    const v4f* __restrict__ vsrc = (const v4f*)video + (size_t)vslice * SLICE4;
    const v4f* __restrict__ nsrc = (const v4f*)noise + (size_t)s      * SLICE4;

    v4f v = __builtin_nontemporal_load(&vsrc[src_pos]);
    v4f n = __builtin_nontemporal_load(&nsrc[src_pos]);

    v4f r;
    #pragma unroll
    for (int k = 0; k < 4; ++k) {
        int cIdx = flip ? (3 - k) : k;          // component reversal for flip
        float e = v[cIdx];
        e = clamp01(e * bf);                    // brightness
        e = clamp01((e - 0.5f) * cf + 0.5f);    // contrast
        e = clamp01(e + n[cIdx] * NOISE_STRENGTH); // noise
        r[k] = e * kp;                          // frame dropout
    }

    __builtin_nontemporal_store(r, (v4f*)out + (size_t)s * SLICE4 + pos);
}

extern "C" void kernel_launch(void* const* d_in, const int* in_sizes, int n_in,
                              void* d_out, int out_size, void* d_ws, size_t ws_size,
                              hipStream_t stream) {
    (void)in_sizes; (void)n_in; (void)out_size; (void)ws_size;
    const float* video     = (const float*)d_in[0];
    const int*   jitter    = (const int*)  d_in[1];
    const float* b_rand    = (const float*)d_in[2];
    const float* c_rand    = (const float*)d_in[3];
    const float* noise     = (const float*)d_in[4];
    const float* flip_rand = (const float*)d_in[5];
    const float* drop_rand = (const float*)d_in[6];

    AugParams* params = (AugParams*)d_ws;

    aug_params_kernel<<<1, 32, 0, stream>>>(jitter, b_rand, c_rand,
                                            flip_rand, drop_rand, params);

    dim3 grid(BLK_X, SLICES);                   // (49, 768)
    aug_main_kernel<<<grid, 256, 0, stream>>>(video, noise, params,
                                              (float*)d_out);
}